// GATMolNode_33947421507890
// MI455X (gfx1250) — compile-verified
//
#include <hip/hip_runtime.h>

// Problem constants (match reference)
#define NNODES   100000
#define NEDGES   1000000
#define DDIM     64
#define HHEADS   4
#define CCH      16
#define NEGS     0.2f
#define BN_EPS   1e-5f

typedef float v2f __attribute__((ext_vector_type(2)));
typedef float v4f __attribute__((ext_vector_type(4)));
typedef float v8f __attribute__((ext_vector_type(8)));

// ---------------------------------------------------------------------------
// float atomic max via signed/unsigned integer punning
// (lowers to GLOBAL_ATOMIC_MAX_I32 / GLOBAL_ATOMIC_MIN_U32 — L2-near atomics)
// ---------------------------------------------------------------------------
__device__ __forceinline__ void atomicMaxF(float* addr, float val) {
  if (val >= 0.0f) {
    atomicMax((int*)addr, __float_as_int(val));
  } else {
    atomicMin((unsigned int*)addr, __float_as_uint(val));
  }
}

// ---------------------------------------------------------------------------
// Self-loop attr mean: degree count + integer attr sums (scatter)
// ---------------------------------------------------------------------------
__global__ void k_zero_deg(int* __restrict__ cnt, int* __restrict__ asum, int n) {
  long i = (long)blockIdx.x * blockDim.x + threadIdx.x;
  if (i >= n) return;
  cnt[i] = 0;
  asum[i * 3 + 0] = 0; asum[i * 3 + 1] = 0; asum[i * 3 + 2] = 0;
}

__global__ void k_deg(const int* __restrict__ dst, const int* __restrict__ eattr,
                      int* __restrict__ cnt, int* __restrict__ asum, int e) {
  long i = (long)blockIdx.x * blockDim.x + threadIdx.x;
  if (i >= e) return;
  int d = dst[i];
  atomicAdd(&cnt[d], 1);
  atomicAdd(&asum[(long)d * 3 + 0], eattr[i * 3 + 0]);
  atomicAdd(&asum[(long)d * 3 + 1], eattr[i * 3 + 1]);
  atomicAdd(&asum[(long)d * 3 + 2], eattr[i * 3 + 2]);
}

__global__ void k_loop_attr(const int* __restrict__ cnt, const int* __restrict__ asum,
                            int* __restrict__ lattr, int n) {
  long i = (long)blockIdx.x * blockDim.x + threadIdx.x;
  if (i >= n) return;
  int c = cnt[i]; if (c < 1) c = 1;
  // attrs are non-negative -> C integer division == floor division
  lattr[i * 3 + 0] = asum[i * 3 + 0] / c;
  lattr[i * 3 + 1] = asum[i * 3 + 1] / c;
  lattr[i * 3 + 2] = asum[i * 3 + 2] / c;
}

// ---------------------------------------------------------------------------
// Atom encoding: h[n,d] = sum_{t=0..8} atom_table_t[x[n,t], d]
// Tables (9 * 64*64 * 4B = 144 KB) are L2-resident gathers.
// ---------------------------------------------------------------------------
__global__ void k_atom_encode(const int* __restrict__ x,
                              const float* __restrict__ t0, const float* __restrict__ t1,
                              const float* __restrict__ t2, const float* __restrict__ t3,
                              const float* __restrict__ t4, const float* __restrict__ t5,
                              const float* __restrict__ t6, const float* __restrict__ t7,
                              const float* __restrict__ t8,
                              float* __restrict__ h, int n) {
  long i = (long)blockIdx.x * blockDim.x + threadIdx.x;
  if (i >= (long)n * DDIM) return;
  int  dd   = (int)(i & (DDIM - 1));
  long node = i >> 6;
  const int* xr = x + node * 9;
  float s = t0[(long)xr[0] * DDIM + dd] + t1[(long)xr[1] * DDIM + dd]
          + t2[(long)xr[2] * DDIM + dd] + t3[(long)xr[3] * DDIM + dd]
          + t4[(long)xr[4] * DDIM + dd] + t5[(long)xr[5] * DDIM + dd]
          + t6[(long)xr[6] * DDIM + dd] + t7[(long)xr[7] * DDIM + dd]
          + t8[(long)xr[8] * DDIM + dd];
  h[i] = s;
}

// ---------------------------------------------------------------------------
// Per-layer scratch init: agg=0, amax=-inf-ish, denom=0, BN accumulators=0
// ---------------------------------------------------------------------------
__global__ void k_layer_init(float* __restrict__ agg, float* __restrict__ amax,
                             float* __restrict__ denom, float* __restrict__ bnsum,
                             float* __restrict__ bnsq, long n64, long n4) {
  long i = (long)blockIdx.x * blockDim.x + threadIdx.x;
  if (i < n64) agg[i] = 0.0f;
  if (i < n4)  { amax[i] = -1.0e30f; denom[i] = 0.0f; }
  if (i < DDIM) { bnsum[i] = 0.0f; bnsq[i] = 0.0f; }
}

// ---------------------------------------------------------------------------
// xt = h @ lin_w via V_WMMA_F32_16X16X4_F32.
// One wave computes a 16-row tile of xt (16x64) = 4 column-tiles x 16 k-steps.
// W (64x64 = 16 KB) staged in LDS once per block; A fragments hoisted into
// registers and reused across the 4 column tiles.
//
// Fragment layouts per CDNA5 ISA (wave32):
//   A 16x4 : lanes 0-15 -> M=lane, K=(0,1) in (v0,v1); lanes 16-31 -> K=(2,3)
//   B 4x16 : lanes 0-15 -> N=lane, K=0 in v0 / K=1 in v1; lanes 16-31 -> K=2/3
//   C 16x16: VGPR v: lanes 0-15 -> M=v, N=lane; lanes 16-31 -> M=v+8, N=lane-16
// ---------------------------------------------------------------------------
__global__ __launch_bounds__(128) void k_gemm_wmma(const float* __restrict__ h,
                                                   const float* __restrict__ W,
                                                   float* __restrict__ xt, int n) {
  __shared__ __align__(16) float sW[DDIM * DDIM];
  {
    const v4f* __restrict__ gw = (const v4f*)W;
    v4f* sw = (v4f*)sW;
    #pragma unroll
    for (int i = threadIdx.x; i < (DDIM * DDIM) / 4; i += 128) sw[i] = gw[i];
  }
  __syncthreads();

  const int wave  = threadIdx.x >> 5;
  const int lane  = threadIdx.x & 31;
  const int mrow  = lane & 15;      // M (for A) / N (for B) within tile
  const int khalf = lane >> 4;      // 0: K pair (0,1), 1: K pair (2,3)

  const long r0 = ((long)blockIdx.x * 4 + wave) * 16;
  if (r0 >= n) return;

  const long arow = r0 + mrow;
  v2f a[16];
  #pragma unroll
  for (int k = 0; k < 16; ++k) {
    const int kb = 4 * k + khalf * 2;
    if (arow < n) {
      a[k].x = h[arow * DDIM + kb];
      a[k].y = h[arow * DDIM + kb + 1];
    } else {
      a[k].x = 0.0f; a[k].y = 0.0f;
    }
  }

  #pragma unroll
  for (int n0 = 0; n0 < DDIM; n0 += 16) {
    v8f c = {0.f, 0.f, 0.f, 0.f, 0.f, 0.f, 0.f, 0.f};
    #pragma unroll
    for (int k = 0; k < 16; ++k) {
      const int kb = 4 * k + khalf * 2;
      v2f b;
      b.x = sW[kb * DDIM + n0 + mrow];
      b.y = sW[(kb + 1) * DDIM + n0 + mrow];
      c = __builtin_amdgcn_wmma_f32_16x16x4_f32(
          /*neg_a=*/false, a[k], /*neg_b=*/false, b,
          /*c_mod=*/(short)0, c, /*reuse_a=*/false, /*reuse_b=*/false);
    }
    #pragma unroll
    for (int v = 0; v < 8; ++v) {
      const long row = r0 + v + khalf * 8;
      if (row < n) xt[row * DDIM + n0 + mrow] = c[v];
    }
  }
}

// ---------------------------------------------------------------------------
// Per-(node, head) attention dots: a_src/a_dst = <xt[n,h,:], att_vec[h,:]>
// ---------------------------------------------------------------------------
__global__ void k_node_att(const float* __restrict__ xt,
                           const float* __restrict__ att_src,
                           const float* __restrict__ att_dst,
                           float* __restrict__ a_src, float* __restrict__ a_dst, int n) {
  long i = (long)blockIdx.x * blockDim.x + threadIdx.x;
  if (i >= (long)n * HHEADS) return;
  int  hh   = (int)(i & (HHEADS - 1));
  long node = i >> 2;
  const float* xp = xt + node * DDIM + hh * CCH;
  float s = 0.0f, d = 0.0f;
  #pragma unroll
  for (int c = 0; c < CCH; ++c) {
    float v = xp[c];
    s += v * att_src[hh * CCH + c];
    d += v * att_dst[hh * CCH + c];
  }
  a_src[i] = s;
  a_dst[i] = d;
}

// ---------------------------------------------------------------------------
// Per-(edge, head): alpha = leaky(a_src[src] + a_dst[dst] + <e, att_edge>)
// + segment max via float atomic max.  Edges [0,E) are real; [E,E+N) are
// self-loops with mean-filled attrs.
// ---------------------------------------------------------------------------
__global__ void k_alpha(const int* __restrict__ src, const int* __restrict__ dst,
                        const int* __restrict__ eattr, const int* __restrict__ lattr,
                        const float* __restrict__ b0, const float* __restrict__ b1,
                        const float* __restrict__ b2, const float* __restrict__ att_edge,
                        const float* __restrict__ a_src, const float* __restrict__ a_dst,
                        float* __restrict__ alpha, float* __restrict__ amax, long total) {
  long i = (long)blockIdx.x * blockDim.x + threadIdx.x;
  if (i >= total) return;
  int  hh = (int)(i & (HHEADS - 1));
  long e  = i >> 2;
  int s, d, t0, t1, t2;
  if (e < NEDGES) {
    s = src[e]; d = dst[e];
    t0 = eattr[e * 3 + 0]; t1 = eattr[e * 3 + 1]; t2 = eattr[e * 3 + 2];
  } else {
    long nn = e - NEDGES;
    s = (int)nn; d = (int)nn;
    t0 = lattr[nn * 3 + 0]; t1 = lattr[nn * 3 + 1]; t2 = lattr[nn * 3 + 2];
  }
  float acc = 0.0f;
  #pragma unroll
  for (int c = 0; c < CCH; ++c) {
    int col = hh * CCH + c;
    float ev = b0[t0 * DDIM + col] + b1[t1 * DDIM + col] + b2[t2 * DDIM + col];
    acc += ev * att_edge[col];
  }
  float al = a_src[(long)s * HHEADS + hh] + a_dst[(long)d * HHEADS + hh] + acc;
  al = (al > 0.0f) ? al : NEGS * al;
  alpha[i] = al;
  atomicMaxF(&amax[(long)d * HHEADS + hh], al);
}

// ---------------------------------------------------------------------------
// ex = exp(alpha - amax[dst]); denom[dst] += ex  (in-place over alpha buffer)
// ---------------------------------------------------------------------------
__global__ void k_expsum(const int* __restrict__ dst, float* __restrict__ alpha_ex,
                         const float* __restrict__ amax, float* __restrict__ denom,
                         long total) {
  long i = (long)blockIdx.x * blockDim.x + threadIdx.x;
  if (i >= total) return;
  int  hh = (int)(i & (HHEADS - 1));
  long e  = i >> 2;
  int d = (e < NEDGES) ? dst[e] : (int)(e - NEDGES);
  float ex = __expf(alpha_ex[i] - amax[(long)d * HHEADS + hh]);
  alpha_ex[i] = ex;
  atomicAdd(&denom[(long)d * HHEADS + hh], ex);
}

// ---------------------------------------------------------------------------
// Message + segment sum: agg[dst,d] += attn * (xt[src,d] + e_enc[d])
// Bond embedding re-gathered (6 KB tables -> L0/L2 hits) instead of storing
// a 280 MB [E+N,64] tensor.
// ---------------------------------------------------------------------------
__global__ void k_message(const int* __restrict__ src, const int* __restrict__ dst,
                          const int* __restrict__ eattr, const int* __restrict__ lattr,
                          const float* __restrict__ b0, const float* __restrict__ b1,
                          const float* __restrict__ b2, const float* __restrict__ xt,
                          const float* __restrict__ ex, const float* __restrict__ denom,
                          float* __restrict__ agg, long total) {
  long i = (long)blockIdx.x * blockDim.x + threadIdx.x;
  if (i >= total) return;
  int  dd = (int)(i & (DDIM - 1));
  long e  = i >> 6;
  int s, d, t0, t1, t2;
  if (e < NEDGES) {
    s = src[e]; d = dst[e];
    t0 = eattr[e * 3 + 0]; t1 = eattr[e * 3 + 1]; t2 = eattr[e * 3 + 2];
  } else {
    long nn = e - NEDGES;
    s = (int)nn; d = (int)nn;
    t0 = lattr[nn * 3 + 0]; t1 = lattr[nn * 3 + 1]; t2 = lattr[nn * 3 + 2];
  }
  int hh = dd >> 4;
  float attn = ex[e * HHEADS + hh] / (denom[(long)d * HHEADS + hh] + 1e-16f);
  float ev = b0[t0 * DDIM + dd] + b1[t1 * DDIM + dd] + b2[t2 * DDIM + dd];
  float msg = attn * (xt[(long)s * DDIM + dd] + ev);
  atomicAdd(&agg[(long)d * DDIM + dd], msg);
}

// ---------------------------------------------------------------------------
// BatchNorm batch stats: per-channel sum / sum-of-squares over N nodes.
// 256 threads = 4 node-lanes x 64 channels; LDS tree + one atomic per channel.
// ---------------------------------------------------------------------------
__global__ __launch_bounds__(256) void k_bn_stats(const float* __restrict__ agg,
                                                  const float* __restrict__ bias,
                                                  float* __restrict__ bnsum,
                                                  float* __restrict__ bnsq, int n) {
  __shared__ float ss[256];
  __shared__ float sq[256];
  int dd  = threadIdx.x & (DDIM - 1);
  int grp = threadIdx.x >> 6;   // 0..3
  float b = bias[dd];
  float s = 0.0f, q = 0.0f;
  for (long node = (long)blockIdx.x * 4 + grp; node < n; node += (long)gridDim.x * 4) {
    float v = agg[node * DDIM + dd] + b;
    s += v; q += v * v;
  }
  ss[threadIdx.x] = s; sq[threadIdx.x] = q;
  __syncthreads();
  if (threadIdx.x < DDIM) {
    s = ss[threadIdx.x] + ss[threadIdx.x + 64] + ss[threadIdx.x + 128] + ss[threadIdx.x + 192];
    q = sq[threadIdx.x] + sq[threadIdx.x + 64] + sq[threadIdx.x + 128] + sq[threadIdx.x + 192];
    atomicAdd(&bnsum[threadIdx.x], s);
    atomicAdd(&bnsq[threadIdx.x], q);
  }
}

// ---------------------------------------------------------------------------
// BN apply (+ optional ReLU): y = (x+bias - mu) * rsqrt(var+eps) * g + b
// ---------------------------------------------------------------------------
__global__ void k_bn_apply(const float* __restrict__ agg, const float* __restrict__ bias,
                           const float* __restrict__ bnsum, const float* __restrict__ bnsq,
                           const float* __restrict__ gamma, const float* __restrict__ beta,
                           float* __restrict__ out, int n, int relu) {
  long i = (long)blockIdx.x * blockDim.x + threadIdx.x;
  if (i >= (long)n * DDIM) return;
  int dd = (int)(i & (DDIM - 1));
  float inv_n = 1.0f / (float)n;
  float mu  = bnsum[dd] * inv_n;
  float var = bnsq[dd] * inv_n - mu * mu;
  float v = agg[i] + bias[dd];
  float y = (v - mu) * rsqrtf(var + BN_EPS) * gamma[dd] + beta[dd];
  if (relu) y = fmaxf(y, 0.0f);
  out[i] = y;
}

// ---------------------------------------------------------------------------
// Host orchestration
// ---------------------------------------------------------------------------
extern "C" void kernel_launch(void* const* d_in, const int* in_sizes, int n_in,
                              void* d_out, int out_size, void* d_ws, size_t ws_size,
                              hipStream_t stream) {
  const int N = NNODES, E = NEDGES;
  const long EN = (long)E + N;

  // Inputs: x, edge_index, edge_attr, then params flattened:
  //   9 atom tables, then per layer: lin_w, att_src, att_dst, att_edge, bias,
  //   bond_tables[0..2], bn_gamma, bn_beta  (10 per layer, 3 layers)
  const int* x     = (const int*)d_in[0];
  const int* ei    = (const int*)d_in[1];
  const int* eattr = (const int*)d_in[2];
  const float* atab[9];
  for (int t = 0; t < 9; ++t) atab[t] = (const float*)d_in[3 + t];
  const int PBASE = 12;

  const int* srcp = ei;
  const int* dstp = ei + E;

  // Carve workspace (256B aligned chunks)
  char* wp = (char*)d_ws;
  auto carve = [&](size_t bytes) -> void* {
    void* p = (void*)wp;
    wp += (bytes + 255) & ~(size_t)255;
    return p;
  };
  float* hA    = (float*)carve((size_t)N * DDIM * 4);
  float* hB    = (float*)carve((size_t)N * DDIM * 4);
  float* hXT   = (float*)carve((size_t)N * DDIM * 4);
  float* hAGG  = (float*)carve((size_t)N * DDIM * 4);
  float* aSrc  = (float*)carve((size_t)N * HHEADS * 4);
  float* aDst  = (float*)carve((size_t)N * HHEADS * 4);
  float* aMax  = (float*)carve((size_t)N * HHEADS * 4);
  float* denom = (float*)carve((size_t)N * HHEADS * 4);
  float* exbuf = (float*)carve((size_t)EN * HHEADS * 4);
  int*   cnt   = (int*)carve((size_t)N * 4);
  int*   asum  = (int*)carve((size_t)N * 3 * 4);
  int*   lattr = (int*)carve((size_t)N * 3 * 4);
  float* bnsum = (float*)carve(DDIM * 4);
  float* bnsq  = (float*)carve(DDIM * 4);

  const int TB = 256;
  auto blocks = [](long total, int tb) -> unsigned { return (unsigned)((total + tb - 1) / tb); };

  // Self-loop attr means
  k_zero_deg<<<blocks(N, TB), TB, 0, stream>>>(cnt, asum, N);
  k_deg<<<blocks(E, TB), TB, 0, stream>>>(dstp, eattr, cnt, asum, E);
  k_loop_attr<<<blocks(N, TB), TB, 0, stream>>>(cnt, asum, lattr, N);

  // Atom embedding encode -> hA
  k_atom_encode<<<blocks((long)N * DDIM, TB), TB, 0, stream>>>(
      x, atab[0], atab[1], atab[2], atab[3], atab[4], atab[5], atab[6], atab[7], atab[8],
      hA, N);

  float* hin  = hA;
  float* hout = hB;
  for (int l = 0; l < 3; ++l) {
    const float* lw    = (const float*)d_in[PBASE + l * 10 + 0];
    const float* psrc  = (const float*)d_in[PBASE + l * 10 + 1];
    const float* pdst  = (const float*)d_in[PBASE + l * 10 + 2];
    const float* pedge = (const float*)d_in[PBASE + l * 10 + 3];
    const float* bias  = (const float*)d_in[PBASE + l * 10 + 4];
    const float* b0    = (const float*)d_in[PBASE + l * 10 + 5];
    const float* b1    = (const float*)d_in[PBASE + l * 10 + 6];
    const float* b2    = (const float*)d_in[PBASE + l * 10 + 7];
    const float* gamma = (const float*)d_in[PBASE + l * 10 + 8];
    const float* beta  = (const float*)d_in[PBASE + l * 10 + 9];

    k_layer_init<<<blocks((long)N * DDIM, TB), TB, 0, stream>>>(
        hAGG, aMax, denom, bnsum, bnsq, (long)N * DDIM, (long)N * HHEADS);

    // xt = h @ lin_w via WMMA (16-row tiles, 4 tiles / 128-thread block)
    const long ntiles = ((long)N + 15) / 16;
    k_gemm_wmma<<<blocks(ntiles, 4), 128, 0, stream>>>(hin, lw, hXT, N);

    k_node_att<<<blocks((long)N * HHEADS, TB), TB, 0, stream>>>(
        hXT, psrc, pdst, aSrc, aDst, N);

    const long totH = EN * HHEADS;
    k_alpha<<<blocks(totH, TB), TB, 0, stream>>>(
        srcp, dstp, eattr, lattr, b0, b1, b2, pedge, aSrc, aDst, exbuf, aMax, totH);
    k_expsum<<<blocks(totH, TB), TB, 0, stream>>>(dstp, exbuf, aMax, denom, totH);

    const long totD = EN * DDIM;
    k_message<<<blocks(totD, TB), TB, 0, stream>>>(
        srcp, dstp, eattr, lattr, b0, b1, b2, hXT, exbuf, denom, hAGG, totD);

    k_bn_stats<<<1024, 256, 0, stream>>>(hAGG, bias, bnsum, bnsq, N);

    float* target = (l == 2) ? (float*)d_out : hout;
    k_bn_apply<<<blocks((long)N * DDIM, TB), TB, 0, stream>>>(
        hAGG, bias, bnsum, bnsq, gamma, beta, target, N, (l < 2) ? 1 : 0);

    float* tmp = hin; hin = hout; hout = tmp;
  }
  (void)in_sizes; (void)n_in; (void)out_size; (void)ws_size;
}